// SAGEConv_59390807769607
// MI455X (gfx1250) — compile-verified
//
#include <hip/hip_runtime.h>
#include <stdint.h>

// copy_u_mean (SAGEConv mean aggregation) on MI455X (gfx1250, wave32).
//
// Pipeline: degree count (atomics) -> hierarchical exclusive scan ->
// CSR fill (fetch-add atomics) -> atomic-free gather+mean.
// x (12.8MB), out (12.8MB), csr (6.4MB) are all L2-resident (192MB L2), so
// the dominant gather phase (~205MB of reads) streams at cache bandwidth.
// Atomics: 3.2M total vs 51.2M for the naive per-feature scatter.

#define FEAT 32
#define WAVE 32

// ---------------------------------------------------------------------------
// wave32 inclusive scan via shuffles (no barriers, no LDS).
// ---------------------------------------------------------------------------
__device__ __forceinline__ uint32_t wave_incl_scan(uint32_t v, int lane) {
    #pragma unroll
    for (int d = 1; d < WAVE; d <<= 1) {
        uint32_t t = __shfl_up(v, d, WAVE);
        if (lane >= d) v += t;
    }
    return v;
}

// ---------------------------------------------------------------------------
// Kernel 1: zero the degree array (d_ws is poisoned with 0xAA by the harness).
// ---------------------------------------------------------------------------
__global__ void k_zero_u32(uint32_t* __restrict__ p, int n) {
    int i = blockIdx.x * blockDim.x + threadIdx.x;
    if (i < n) p[i] = 0u;
}

// ---------------------------------------------------------------------------
// Kernel 2: in-degree histogram. 1.6M non-returning global atomics.
// ---------------------------------------------------------------------------
__global__ void k_count_deg(const int* __restrict__ dst,
                            uint32_t* __restrict__ deg, int E) {
    int e = blockIdx.x * blockDim.x + threadIdx.x;
    if (e < E) atomicAdd(&deg[dst[e]], 1u);
}

// ---------------------------------------------------------------------------
// Scan S1: per-block (1024-elem chunk) exclusive scan, barrier-light:
// shfl scan per wave + one LDS hop for the 32 wave sums (2 barriers total).
// Writes chunk-local exclusive values and the block total.
// ---------------------------------------------------------------------------
__global__ __launch_bounds__(1024) void k_scan_block(
        const uint32_t* __restrict__ in,
        uint32_t* __restrict__ out_raw,
        uint32_t* __restrict__ bsum, int n) {
    __shared__ uint32_t wsum[32], wex[32];
    const int tid  = threadIdx.x;
    const int lane = tid & (WAVE - 1);
    const int wv   = tid >> 5;
    const int i    = blockIdx.x * 1024 + tid;

    const uint32_t v    = (i < n) ? in[i] : 0u;
    const uint32_t incl = wave_incl_scan(v, lane);
    if (lane == WAVE - 1) wsum[wv] = incl;
    __syncthreads();
    if (wv == 0) {
        const uint32_t s     = wsum[lane];
        const uint32_t sincl = wave_incl_scan(s, lane);
        wex[lane] = sincl - s;                       // exclusive wave base
        if (lane == WAVE - 1) bsum[blockIdx.x] = sincl;  // block total
    }
    __syncthreads();
    if (i < n) out_raw[i] = wex[wv] + incl - v;      // chunk-local exclusive
}

// ---------------------------------------------------------------------------
// Scan S2: exclusive scan of the block sums (nb <= 1024), single block.
// ---------------------------------------------------------------------------
__global__ __launch_bounds__(1024) void k_scan_bsum(
        const uint32_t* __restrict__ bsum,
        uint32_t* __restrict__ bpre, int nb) {
    __shared__ uint32_t wsum[32], wex[32];
    const int tid  = threadIdx.x;
    const int lane = tid & (WAVE - 1);
    const int wv   = tid >> 5;

    const uint32_t v    = (tid < nb) ? bsum[tid] : 0u;
    const uint32_t incl = wave_incl_scan(v, lane);
    if (lane == WAVE - 1) wsum[wv] = incl;
    __syncthreads();
    if (wv == 0) {
        const uint32_t s     = wsum[lane];
        const uint32_t sincl = wave_incl_scan(s, lane);
        wex[lane] = sincl - s;
    }
    __syncthreads();
    if (tid < nb) bpre[tid] = wex[wv] + incl - v;
}

// ---------------------------------------------------------------------------
// Scan S3: finalize offs (global exclusive scan) and cur (fill cursor copy).
// ---------------------------------------------------------------------------
__global__ __launch_bounds__(1024) void k_add_base(
        uint32_t* __restrict__ offs,
        uint32_t* __restrict__ cur,
        const uint32_t* __restrict__ bpre, int n) {
    const int i = blockIdx.x * 1024 + threadIdx.x;
    if (i < n) {
        const uint32_t o = offs[i] + bpre[blockIdx.x];
        offs[i] = o;
        cur[i]  = o;
    }
}

// ---------------------------------------------------------------------------
// Kernel 4: CSR fill. Each edge claims a unique slot in its destination's
// segment via returning atomic fetch-add. 1.6M atomics.
// ---------------------------------------------------------------------------
__global__ void k_fill_csr(const int* __restrict__ src,
                           const int* __restrict__ dst,
                           uint32_t* __restrict__ cur,
                           uint32_t* __restrict__ csr, int E) {
    int e = blockIdx.x * blockDim.x + threadIdx.x;
    if (e < E) {
        const uint32_t p = atomicAdd(&cur[dst[e]], 1u);
        csr[p] = (uint32_t)src[e];
    }
}

// ---------------------------------------------------------------------------
// Kernel 5: atomic-free gather + mean.
// wave32 layout: 8 lanes per node (lane loads float4 of the 32-float row), so
// each edge row is one coalesced 128B global_load_b128 per 8-lane group;
// 4 nodes per wave. Unroll 4 edges (independent b128 loads the compiler can
// clause) and prefetch the random row 8 edges ahead (global_prefetch_b8).
// ---------------------------------------------------------------------------
__global__ __launch_bounds__(256) void k_aggregate(
        const float* __restrict__ x,
        const uint32_t* __restrict__ csr,
        const uint32_t* __restrict__ offs,
        const uint32_t* __restrict__ deg,
        float* __restrict__ out, int n) {
    const int gtid = blockIdx.x * blockDim.x + threadIdx.x;
    const int node = gtid >> 3;       // 8 lanes per node
    const int sub  = (gtid & 7) * 4;  // feature chunk [sub, sub+4)
    if (node >= n) return;

    const uint32_t start = offs[node];
    const uint32_t d     = deg[node];

    float a0 = 0.f, a1 = 0.f, a2 = 0.f, a3 = 0.f;

    uint32_t i = 0;
    for (; i + 4 <= d; i += 4) {
        const uint32_t s0 = csr[start + i];
        const uint32_t s1 = csr[start + i + 1];
        const uint32_t s2 = csr[start + i + 2];
        const uint32_t s3 = csr[start + i + 3];
        if (i + 8 < d) {   // hide random-gather latency: prefetch 8 ahead
            const uint32_t sp = csr[start + i + 8];
            __builtin_prefetch(&x[(size_t)sp * FEAT + sub], 0, 0);
        }
        const float4 v0 = *(const float4*)&x[(size_t)s0 * FEAT + sub];
        const float4 v1 = *(const float4*)&x[(size_t)s1 * FEAT + sub];
        const float4 v2 = *(const float4*)&x[(size_t)s2 * FEAT + sub];
        const float4 v3 = *(const float4*)&x[(size_t)s3 * FEAT + sub];
        a0 += v0.x; a1 += v0.y; a2 += v0.z; a3 += v0.w;
        a0 += v1.x; a1 += v1.y; a2 += v1.z; a3 += v1.w;
        a0 += v2.x; a1 += v2.y; a2 += v2.z; a3 += v2.w;
        a0 += v3.x; a1 += v3.y; a2 += v3.z; a3 += v3.w;
    }
    for (; i < d; ++i) {
        const uint32_t s0 = csr[start + i];
        const float4 v0 = *(const float4*)&x[(size_t)s0 * FEAT + sub];
        a0 += v0.x; a1 += v0.y; a2 += v0.z; a3 += v0.w;
    }

    // Reference: msg_sum / max(deg, 1). IEEE f32 divide; deg==0 -> 0.
    const float dd = (float)(d > 0u ? d : 1u);
    float4 r;
    r.x = a0 / dd; r.y = a1 / dd; r.z = a2 / dd; r.w = a3 / dd;
    *(float4*)&out[(size_t)node * FEAT + sub] = r;
}

// ---------------------------------------------------------------------------
// Launcher
// ---------------------------------------------------------------------------
extern "C" void kernel_launch(void* const* d_in, const int* in_sizes, int n_in,
                              void* d_out, int out_size, void* d_ws, size_t ws_size,
                              hipStream_t stream) {
    const float* x   = (const float*)d_in[0];
    const int*   src = (const int*)d_in[1];   // JAX default x64-disabled -> int32
    const int*   dst = (const int*)d_in[2];
    float*       out = (float*)d_out;

    const int n = in_sizes[0] / FEAT;         // 100000 nodes
    const int E = in_sizes[1];                // 1600000 edges
    const int nb = (n + 1023) / 1024;         // scan blocks (98)

    // Scratch layout (256B-aligned slabs): deg | offs | cur | bsum | bpre | csr
    uint8_t* ws = (uint8_t*)d_ws;
    const size_t slab  = (((size_t)n * 4u) + 255u) & ~(size_t)255u;
    const size_t bslab = (((size_t)nb * 4u) + 255u) & ~(size_t)255u;
    uint32_t* deg  = (uint32_t*)(ws);
    uint32_t* offs = (uint32_t*)(ws + slab);
    uint32_t* cur  = (uint32_t*)(ws + 2 * slab);
    uint32_t* bsum = (uint32_t*)(ws + 3 * slab);
    uint32_t* bpre = (uint32_t*)(ws + 3 * slab + bslab);
    uint32_t* csr  = (uint32_t*)(ws + 3 * slab + 2 * bslab);  // E u32 = 6.4MB

    const int blk = 256;

    k_zero_u32  <<<(n + blk - 1) / blk, blk, 0, stream>>>(deg, n);
    k_count_deg <<<(E + blk - 1) / blk, blk, 0, stream>>>(dst, deg, E);

    // Hierarchical exclusive scan: deg -> offs (and cur).
    k_scan_block<<<nb,   1024, 0, stream>>>(deg, offs, bsum, n);
    k_scan_bsum <<<1,    1024, 0, stream>>>(bsum, bpre, nb);
    k_add_base  <<<nb,   1024, 0, stream>>>(offs, cur, bpre, n);

    k_fill_csr  <<<(E + blk - 1) / blk, blk, 0, stream>>>(src, dst, cur, csr, E);

    const int agg_threads = n * 8;            // 8 lanes per node
    k_aggregate <<<(agg_threads + blk - 1) / blk, blk, 0, stream>>>(
        x, csr, offs, deg, out, n);
}